// GATLayer_24249385353673
// MI455X (gfx1250) — compile-verified
//
#include <hip/hip_runtime.h>
#include <math.h>

// GAT layer for MI455X (gfx1250, wave32).
// B=8, N=1024, D_IN=256, D_OUT=256, H=8, HD=32, ALPHA=0.2
//
// Memory-bound: adj mask (int64) = 64MB dominates -> packed to 1MB bitmask.
// Both GEMMs use V_WMMA_F32_16X16X4_F32 (keeps reference f32 precision; the
// problem is memory/TRANS-bound so low-precision WMMA buys nothing).
// B operands of both GEMMs are pre-transposed so every B-fragment is a single
// global_load_b64 (two K-consecutive values) instead of two strided b32s.

typedef __attribute__((ext_vector_type(2))) float v2f;
typedef __attribute__((ext_vector_type(8))) float v8f;

#define NN 1024
#define DD 256
#define HH_ 8
#define BB 8

// ---------------------------------------------------------------------------
// Kernel 0: WT[n][k] = W[k][n]  (256x256, makes wh_gemm B-frags contiguous)
// ---------------------------------------------------------------------------
__global__ __launch_bounds__(256) void wt_kernel(const float* __restrict__ W,
                                                 float* __restrict__ WT) {
  const int t = blockIdx.x * blockDim.x + threadIdx.x;  // t = n*256 + k
  const int k = t & (DD - 1);
  const int n = t >> 8;
  WT[t] = W[k * DD + n];
}

// ---------------------------------------------------------------------------
// Kernel 1: Wh = h @ W   (8192x256 @ 256x256, f32 WMMA 16x16x4)
// One wave per 16x16 output tile. A-frag (ISA 7.12.2, 32-bit A 16x4):
//   lanes 0-15: M=lane, VGPR0=K+0, VGPR1=K+1 ; lanes 16-31: M=lane-16, K+2/K+3
// B-frag (4x16): N=lane%16; VGPR0/1 = rows K+(lane<16?0:2), +1  -> from WT
// these are consecutive: one b64 load.
// ---------------------------------------------------------------------------
__global__ __launch_bounds__(32) void wh_gemm(const float* __restrict__ h,
                                              const float* __restrict__ WT,
                                              float* __restrict__ Wh) {
  const int lane = threadIdx.x;
  const int row0 = blockIdx.x * 16;
  const int col0 = blockIdx.y * 16;
  const int m16  = lane & 15;
  const int kpar = (lane < 16) ? 0 : 2;

  const float* arow = h + (row0 + m16) * DD;            // one h row per lane
  const float* bcol = WT + (col0 + m16) * DD;           // one W^T row per lane
  v8f acc = {};

  #pragma unroll 4
  for (int k = 0; k < DD; k += 4) {
    v2f a = *(const v2f*)(arow + k + kpar);             // global_load_b64
    v2f b = *(const v2f*)(bcol + k + kpar);             // global_load_b64
    acc = __builtin_amdgcn_wmma_f32_16x16x4_f32(false, a, false, b,
                                                (short)0, acc, false, false);
  }

  // C/D layout: lane holds col N=lane%16, rows r+(lane<16?0:8) in VGPR r.
  const int rbase = (lane < 16) ? 0 : 8;
  float* outp = Wh + (long long)(row0 + rbase) * DD + col0 + m16;
  #pragma unroll
  for (int r = 0; r < 8; ++r)
    outp[(long long)r * DD] = acc[r];
}

// ---------------------------------------------------------------------------
// Kernel 2: e1/e2 projections + per-head transposed copy
//   e1[b,h,n] = sum_d Wh[b,n,h*32+d] * a[d]      (a1)
//   e2[b,h,n] = sum_d Wh[b,n,h*32+d] * a[32+d]   (a2)
//   VT[b,h,d,n] = Wh[b,n,h*32+d]                 (V^T for the P@V WMMA)
// ---------------------------------------------------------------------------
__global__ __launch_bounds__(256) void e12_vt_kernel(const float* __restrict__ Wh,
                                                     const float* __restrict__ a,
                                                     float* __restrict__ e1,
                                                     float* __restrict__ e2,
                                                     float* __restrict__ VT) {
  const int t = blockIdx.x * blockDim.x + threadIdx.x;   // (b*H + h)*N + n
  const int b  = t >> 13;          // H*N = 8192
  const int hh = (t >> 10) & 7;
  const int n  = t & (NN - 1);
  const int bh = (b << 3) | hh;
  const float* wrow = Wh + (long long)(b * NN + n) * DD + hh * 32;
  float* vtcol = VT + (long long)bh * 32 * NN + n;       // stride NN per d
  float s1 = 0.f, s2 = 0.f;
  #pragma unroll
  for (int d = 0; d < 32; ++d) {
    float v = wrow[d];
    s1 += v * a[d];
    s2 += v * a[32 + d];
    vtcol[(long long)d * NN] = v;
  }
  e1[t] = s1;
  e2[t] = s2;
}

// ---------------------------------------------------------------------------
// Kernel 3: pack int64 adjacency (64MB) into a bitmask (1MB).
// bits[b][i][w] bit j  <=>  adj[b][i][w*32+j] != 0
// ---------------------------------------------------------------------------
__global__ __launch_bounds__(256) void pack_adj(const long long* __restrict__ adj,
                                                unsigned* __restrict__ bits) {
  const int t = blockIdx.x * blockDim.x + threadIdx.x;   // b*32768 + i*32 + w
  const int w = t & 31;
  const int i = (t >> 5) & (NN - 1);
  const int b = t >> 15;
  const long long* row = adj + ((long long)(b * NN + i)) * NN + w * 32;
  unsigned m = 0u;
  #pragma unroll 8
  for (int j = 0; j < 32; ++j)
    if (row[j] != 0) m |= (1u << j);
  bits[t] = m;
}

// ---------------------------------------------------------------------------
// Kernel 4: fused mask + leakyrelu + online softmax + (attn @ Wh).
// One wave per (b,h, 16-row i-tile). j in chunks of 16; per chunk:
//  - each lane computes the 8 logits landing in its A-frag slots
//    (lane<16: j in {4q,4q+1}, lane>=16: j in {4q+2,4q+3})
//  - flash-style running max/sum; partner-lane (xor 16) reduction gives row
//    stats; accumulator rescale is vote-skipped when no row max moved
//  - 4 K-steps x 2 N-tiles of v_wmma_f32_16x16x4_f32; B-frags are b64 loads
//    from VT (K-consecutive), L2-resident (128KB per (b,h), shared by 64 blocks)
// ---------------------------------------------------------------------------
__global__ __launch_bounds__(32) void gat_attn(const float* __restrict__ VT,
                                               const float* __restrict__ e1,
                                               const float* __restrict__ e2,
                                               const unsigned* __restrict__ bits,
                                               float* __restrict__ out) {
  const int lane = threadIdx.x;
  const int i0 = blockIdx.x * 16;
  const int bh = blockIdx.y;            // b*H + h
  const int b  = bh >> 3;
  const int hh = bh & 7;
  const int m16  = lane & 15;
  const int jpar = (lane < 16) ? 0 : 2;
  const int statrow = i0 + m16;         // row whose softmax stats this lane owns

  const float  e1i  = e1[bh * NN + statrow];
  const float* e2p  = e2 + bh * NN;
  const unsigned* brow = bits + (b * NN + statrow) * 32;
  // V^T rows for this lane's two B-frag columns (d = ntile*16 + m16)
  const float* vt0 = VT + ((long long)bh * 32 + m16) * NN;
  const float* vt1 = vt0 + 16 * NN;

  float m = -INFINITY, l = 0.f;
  v8f acc0 = {}, acc1 = {};

  for (int jc = 0; jc < NN; jc += 16) {
    const unsigned word = brow[jc >> 5];
    const unsigned msk  = (word >> (jc & 16)) & 0xFFFFu;

    float p[8];
    float cmax = -INFINITY;
    #pragma unroll
    for (int q = 0; q < 4; ++q) {
      v2f e2v = *(const v2f*)(e2p + jc + 4 * q + jpar);   // even offset: b64
      #pragma unroll
      for (int s = 0; s < 2; ++s) {
        const int j = 4 * q + jpar + s;
        float e = e1i + e2v[s];
        e = (e >= 0.f) ? e : 0.2f * e;                    // leaky relu
        if (!((msk >> j) & 1u)) e = -INFINITY;            // adjacency mask
        p[2 * q + s] = e;
        cmax = fmaxf(cmax, e);
      }
    }
    cmax = fmaxf(cmax, __shfl_xor(cmax, 16, 32));         // full 16-wide row max
    const float newm  = fmaxf(m, cmax);
    const float scale = (newm > -3.0e38f) ? __expf(m - newm) : 1.0f;

    float psum = 0.f;
    #pragma unroll
    for (int t = 0; t < 8; ++t) {
      const float e  = p[t];
      const float pe = (e > -3.0e38f) ? __expf(e - newm) : 0.f;
      p[t] = pe;
      psum += pe;
    }
    psum += __shfl_xor(psum, 16, 32);
    l = l * scale + psum;
    m = newm;

    // rescale accumulators only if some row's max moved (uniform branch:
    // scalar condition from vote, EXEC stays all-ones for the WMMAs below)
    if (__any(scale < 1.0f)) {
      const int rb = (lane < 16) ? 0 : 8;   // row M's scale lives at lane M
      #pragma unroll
      for (int r = 0; r < 8; ++r) {
        const float sc = __shfl(scale, r + rb, 32);
        acc0[r] *= sc;
        acc1[r] *= sc;
      }
    }

    // P @ V : A = P(16x16) as 4 K-chunks, B from V^T (b64, K-consecutive)
    #pragma unroll
    for (int q = 0; q < 4; ++q) {
      v2f a;
      a.x = p[2 * q];
      a.y = p[2 * q + 1];
      const int krow = jc + 4 * q + jpar;                 // even -> b64 ok
      v2f b0 = *(const v2f*)(vt0 + krow);
      v2f b1 = *(const v2f*)(vt1 + krow);
      acc0 = __builtin_amdgcn_wmma_f32_16x16x4_f32(false, a, false, b0,
                                                   (short)0, acc0, false, false);
      acc1 = __builtin_amdgcn_wmma_f32_16x16x4_f32(false, a, false, b1,
                                                   (short)0, acc1, false, false);
    }
  }

  // normalize by row sum l and store: out[b, i0+M, h*32 + ntile*16 + m16]
  const int rb = (lane < 16) ? 0 : 8;
  float* obase = out + ((long long)(b * NN + i0 + rb)) * DD + hh * 32 + m16;
  #pragma unroll
  for (int r = 0; r < 8; ++r) {
    const float lr  = __shfl(l, r + rb, 32);
    const float inv = 1.0f / lr;
    obase[(long long)r * DD]      = acc0[r] * inv;
    obase[(long long)r * DD + 16] = acc1[r] * inv;
  }
}

// ---------------------------------------------------------------------------
extern "C" void kernel_launch(void* const* d_in, const int* in_sizes, int n_in,
                              void* d_out, int out_size, void* d_ws, size_t ws_size,
                              hipStream_t stream) {
  const float*     h_in = (const float*)d_in[0];      // (8,1024,256) f32
  const long long* adj  = (const long long*)d_in[1];  // (8,1024,1024) i64
  const float*     W    = (const float*)d_in[2];      // (256,256) f32
  const float*     a    = (const float*)d_in[3];      // (64,) f32
  float*           outp = (float*)d_out;              // (8,1024,256) f32

  // workspace layout (floats): Wh | VT | e1 | e2 | WT | bits(u32)
  float* ws = (float*)d_ws;
  float* Wh = ws;                                       // 2,097,152 f (8MB)
  float* VT = Wh + (size_t)BB * NN * DD;                // 2,097,152 f (8MB)
  float* e1 = VT + (size_t)BB * NN * DD;                // 65,536 f
  float* e2 = e1 + (size_t)BB * HH_ * NN;               // 65,536 f
  float* WT = e2 + (size_t)BB * HH_ * NN;               // 65,536 f
  unsigned* bits = (unsigned*)(WT + (size_t)DD * DD);   // 262,144 u32 (1MB)

  // 0) transpose W (256KB) so wh_gemm B-frags are b64
  wt_kernel<<<(DD * DD) / 256, 256, 0, stream>>>(W, WT);

  // 1) Wh = h @ W   (wave per 16x16 tile: 512 x 16 tiles)
  wh_gemm<<<dim3(BB * NN / 16, DD / 16), 32, 0, stream>>>(h_in, WT, Wh);

  // 2) e1/e2 projections + per-head V^T copy (B*H*N = 65536 threads)
  e12_vt_kernel<<<(BB * HH_ * NN) / 256, 256, 0, stream>>>(Wh, a, e1, e2, VT);

  // 3) pack int64 adjacency into bitmask (B*N*32 = 262144 threads)
  pack_adj<<<(BB * NN * 32) / 256, 256, 0, stream>>>(adj, bits);

  // 4) fused masked softmax + aggregation (wave per (b,h,i-tile): 64 x 64)
  gat_attn<<<dim3(NN / 16, BB * HH_), 32, 0, stream>>>(VT, e1, e2, bits, outp);
}